// CanonicalizationWrapper_55456617726381
// MI455X (gfx1250) — compile-verified
//
#include <hip/hip_runtime.h>
#include <cstdint>
#include <cstddef>

#define B_DIM 128
#define N_PTS 65536
#define EPSV  1e-8f
#define TILE_PTS 1024            // points staged per workgroup in k_moments

typedef __attribute__((ext_vector_type(2))) float        v2f;
typedef __attribute__((ext_vector_type(8))) float        v8f;
typedef __attribute__((ext_vector_type(4))) unsigned int v4u;
typedef __attribute__((ext_vector_type(4))) int          v4i;
typedef __attribute__((ext_vector_type(8))) int          v8i;

struct BatchStats {            // 64 floats = 256 B per batch
  float M[16];                 // 4x4 moment matrix: M[m][n]; M[d][e]=sum p_d p_e, M[d][3]=sum p_d, M[3][3]=N
  float mean[3];
  float maxnorm;               // holds max ||x-mean||^2 (as float bits via atomicMax), later sqrt'ed
  float R[9];                  // row-major R[d][e]: canonical_e = sum_d pc_d * R[d][e]
  float skew[3];
  float s[3];
  float invmn;
  float pad[28];
};

// ---------------------------------------------------------------- zero stats
__global__ void k_zero(BatchStats* st) {
  float* f = (float*)st;
  int total = B_DIM * (int)(sizeof(BatchStats) / 4);
  for (int i = blockIdx.x * blockDim.x + threadIdx.x; i < total;
       i += gridDim.x * blockDim.x)
    f[i] = 0.0f;
}

// ------------------------------------------------- pass 1: moments via WMMA
// One TDM descriptor per workgroup DMAs a 12 KB point tile into LDS
// (TENSOR_LOAD_TO_LDS, tracked by TENSORcnt, no VALU/VMEM occupancy), then
// WMMA fragments are fed from LDS. A(16x4): rows 0..2 = coords of 4 points,
// row 3 = ones; B mirrors A. D += A*B accumulates sum(p p^T) in
// D[0..2][0..2], sum(p) in D[0..2][3], and the count in D[3][3].
__global__ void __launch_bounds__(256) k_moments(const float* __restrict__ x,
                                                 BatchStats* st) {
  __shared__ float tile[TILE_PTS * 3];     // 12 KB

  const int b      = blockIdx.y;
  const int bx     = blockIdx.x;           // 64 blocks per batch
  const int tid    = threadIdx.x;
  const int lane   = tid & 31;
  const int waveId = tid >> 5;             // 0..7
  const int m      = lane & 15;            // matrix row this lane feeds
  const int half   = lane >> 4;            // selects K pair {0,1} or {2,3}

  const float* src = x + ((size_t)b * N_PTS + (size_t)bx * TILE_PTS) * 3;

  // ---- TDM: 1-D tile, 3072 dwords (12 KB), global -> LDS
  if (waveId == 0) {
    const uint64_t ga = (uint64_t)(uintptr_t)src;          // 57-bit global addr
    const uint32_t la = (uint32_t)(uintptr_t)(&tile[0]);   // LDS byte offset
    v4u g0;
    g0.x = 1u;                                   // count=1 valid descriptor
    g0.y = la;                                   // lds_addr
    g0.z = (uint32_t)(ga & 0xffffffffu);         // global_addr[31:0]
    g0.w = ((uint32_t)(ga >> 32) & 0x01ffffffu)  // global_addr[56:32]
           | (2u << 30);                         // type=2 ("image")
    v8i g1;
    g1[0] = (int)(2u << 16);                     // data_size=2 (4 bytes)
    g1[1] = (int)(3072u << 16);                  // tensor_dim0[15:0] in bits 63:48
    g1[2] = 0;                                   // tensor_dim0[31:16]=0, tensor_dim1=0
    g1[3] = (int)(3072u << 16);                  // tile_dim0=3072 in bits 127:112
    g1[4] = 0;                                   // tile_dim1=0 (unused), tile_dim2=0
    g1[5] = 3072;                                // tensor_dim0_stride[31:0]
    g1[6] = 0;
    g1[7] = 0;
    v4i gz4 = {0, 0, 0, 0};
    v8i gz8 = {0, 0, 0, 0, 0, 0, 0, 0};
    __builtin_amdgcn_tensor_load_to_lds(g0, g1, gz4, gz4, gz8, 0);
    __builtin_amdgcn_s_wait_tensorcnt(0);
  }
  __syncthreads();

  // ---- WMMA accumulation over the 256 four-point groups in the tile.
  // Lane roles are loop-invariant: clamp the LDS offset and select with
  // cndmask (no per-iteration exec-mask churn).
  const bool  isCoord = (m < 3);
  const float onesVal = (m == 3) ? 1.f : 0.f;
  const int   mc      = isCoord ? m : 0;
  const int   k0 = 2 * half, k1 = 2 * half + 1;
  const int   offX = 3 * k0 + mc;            // dword offsets within a group
  const int   offY = 3 * k1 + mc;

  v8f acc0 = {0.f, 0.f, 0.f, 0.f, 0.f, 0.f, 0.f, 0.f};
  v8f acc1 = acc0;
  for (int g = waveId * 2; g < 256; g += 16) {         // 16 iterations
    const int o0 = g * 12, o1 = o0 + 12;
    float l0x = tile[o0 + offX], l0y = tile[o0 + offY];
    float l1x = tile[o1 + offX], l1y = tile[o1 + offY];
    v2f a0; a0.x = isCoord ? l0x : onesVal; a0.y = isCoord ? l0y : onesVal;
    v2f a1; a1.x = isCoord ? l1x : onesVal; a1.y = isCoord ? l1y : onesVal;
    acc0 = __builtin_amdgcn_wmma_f32_16x16x4_f32(
        false, a0, false, a0, (short)0, acc0, false, false);
    acc1 = __builtin_amdgcn_wmma_f32_16x16x4_f32(
        false, a1, false, a1, (short)0, acc1, false, false);
  }
  acc0 = acc0 + acc1;

  // C/D layout: VGPR r, lanes 0-15 -> D[r][lane]. We need D[0..3][0..3].
  if (lane < 4) {
    atomicAdd(&st[b].M[0 * 4 + lane], acc0[0]);
    atomicAdd(&st[b].M[1 * 4 + lane], acc0[1]);
    atomicAdd(&st[b].M[2 * 4 + lane], acc0[2]);
    atomicAdd(&st[b].M[3 * 4 + lane], acc0[3]);
  }
}

// ------------------------------------------------------------- mean (tiny)
__global__ void k_mean(BatchStats* st) {
  int b = blockIdx.x * blockDim.x + threadIdx.x;
  if (b < B_DIM) {
    const float invN = 1.0f / (float)N_PTS;
    for (int d = 0; d < 3; ++d) st[b].mean[d] = st[b].M[d * 4 + 3] * invN;
  }
}

// ----------------------------------------------- pass 2: max ||x - mean||^2
// Stage the 3 KB point tile with coalesced async global->LDS b128 copies
// (ASYNCcnt), then read stride-3 from LDS (coprime with 64 banks).
__global__ void __launch_bounds__(256) k_maxnorm(const float* __restrict__ x,
                                                 BatchStats* st) {
  __shared__ float ptile[256 * 3];   // 3 KB
  __shared__ float red[256];
  const int tid = threadIdx.x;
  const int pt  = blockIdx.x * blockDim.x + tid;          // over B*N
  const int b   = pt >> 16;                               // N = 65536

  const float* src = x + (size_t)(blockIdx.x * blockDim.x) * 3;
  const uint32_t ldsBase = (uint32_t)(uintptr_t)(&ptile[0]);
  if (tid < 192) {                                        // 192 x 16B = 3 KB
    const uint64_t ga = (uint64_t)(uintptr_t)((const char*)src + (size_t)tid * 16);
    asm volatile("global_load_async_to_lds_b128 %0, %1, off"
                 :: "v"(ldsBase + (uint32_t)tid * 16u), "v"(ga) : "memory");
  }
  asm volatile("s_wait_asynccnt 0x0" ::: "memory");
  __syncthreads();

  float dx = ptile[tid * 3 + 0] - st[b].mean[0];
  float dy = ptile[tid * 3 + 1] - st[b].mean[1];
  float dz = ptile[tid * 3 + 2] - st[b].mean[2];
  red[tid] = dx * dx + dy * dy + dz * dz;
  __syncthreads();
  for (int s = 128; s > 0; s >>= 1) {
    if (tid < s) red[tid] = fmaxf(red[tid], red[tid + s]);
    __syncthreads();
  }
  if (tid == 0)
    atomicMax((unsigned int*)&st[b].maxnorm, __float_as_uint(red[0]));
}

// ---------------------------- pass 3: 3x3 symmetric eigen (Jacobi) per batch
__global__ void k_eigen(BatchStats* st) {
  int b = blockIdx.x * blockDim.x + threadIdx.x;
  if (b >= B_DIM) return;
  BatchStats& S = st[b];

  float mn  = sqrtf(S.maxnorm);
  mn        = fmaxf(mn, EPSV);
  float inv = 1.0f / mn;
  S.maxnorm = mn;
  S.invmn   = inv;

  const float Nf = (float)N_PTS;
  float a[3][3];
  for (int d = 0; d < 3; ++d)
    for (int e = 0; e < 3; ++e) {
      float cde = (S.M[d * 4 + e] - Nf * S.mean[d] * S.mean[e]) * inv * inv /
                  (Nf - 1.0f);
      a[d][e] = cde + ((d == e) ? EPSV : 0.0f);
    }

  float V[3][3] = {{1, 0, 0}, {0, 1, 0}, {0, 0, 1}};
  const int PP[3] = {0, 0, 1}, QQ[3] = {1, 2, 2};
  for (int it = 0; it < 30; ++it) {
    int p = PP[it % 3], q = QQ[it % 3];
    float apq = a[p][q];
    if (fabsf(apq) > 1e-20f) {
      float theta = (a[q][q] - a[p][p]) / (2.0f * apq);
      float t = ((theta >= 0.f) ? 1.f : -1.f) /
                (fabsf(theta) + sqrtf(theta * theta + 1.0f));
      float c = 1.0f / sqrtf(t * t + 1.0f);
      float sn = t * c;
      for (int k = 0; k < 3; ++k) {
        float akp = a[k][p], akq = a[k][q];
        a[k][p] = c * akp - sn * akq;
        a[k][q] = sn * akp + c * akq;
      }
      for (int k = 0; k < 3; ++k) {
        float apk = a[p][k], aqk = a[q][k];
        a[p][k] = c * apk - sn * aqk;
        a[q][k] = sn * apk + c * aqk;
      }
      for (int k = 0; k < 3; ++k) {
        float vkp = V[k][p], vkq = V[k][q];
        V[k][p] = c * vkp - sn * vkq;
        V[k][q] = sn * vkp + c * vkq;
      }
    }
  }

  // sort eigenpairs descending (eigh ascending, then reversed)
  float ev[3] = {a[0][0], a[1][1], a[2][2]};
  int ord[3] = {0, 1, 2};
  for (int i = 0; i < 3; ++i)
    for (int j = 0; j < 2; ++j)
      if (ev[ord[j]] < ev[ord[j + 1]]) { int t = ord[j]; ord[j] = ord[j + 1]; ord[j + 1] = t; }

  float Rm[3][3];
  for (int e = 0; e < 3; ++e) {
    int col = ord[e];
    int mi = 0; float mv = fabsf(V[0][col]);
    for (int r = 1; r < 3; ++r) {
      float av = fabsf(V[r][col]);
      if (av > mv) { mv = av; mi = r; }             // strict > keeps first (argmax)
    }
    float sg = (V[mi][col] >= 0.0f) ? 1.0f : -1.0f; // sign(0) -> +1
    for (int r = 0; r < 3; ++r) Rm[r][e] = V[r][col] * sg;
  }
  float det = Rm[0][0] * (Rm[1][1] * Rm[2][2] - Rm[1][2] * Rm[2][1]) -
              Rm[0][1] * (Rm[1][0] * Rm[2][2] - Rm[1][2] * Rm[2][0]) +
              Rm[0][2] * (Rm[1][0] * Rm[2][1] - Rm[1][1] * Rm[2][0]);
  if (det < 0.0f)
    for (int r = 0; r < 3; ++r) Rm[r][2] = -Rm[r][2];
  for (int d = 0; d < 3; ++d)
    for (int e = 0; e < 3; ++e) S.R[d * 3 + e] = Rm[d][e];
}

// ------------------------- pass 4: canonical = pc*R, accumulate skew = sum c^3
__global__ void __launch_bounds__(256) k_transform(const float* __restrict__ x,
                                                   BatchStats* st,
                                                   float* __restrict__ canon) {
  __shared__ float red[256];
  const int pt = blockIdx.x * blockDim.x + threadIdx.x;
  const int b  = pt >> 16;
  const BatchStats& S = st[b];
  const float* p = x + (size_t)pt * 3;
  float inv = S.invmn;
  float p0 = (p[0] - S.mean[0]) * inv;
  float p1 = (p[1] - S.mean[1]) * inv;
  float p2 = (p[2] - S.mean[2]) * inv;
  float c[3];
  c[0] = p0 * S.R[0] + p1 * S.R[3] + p2 * S.R[6];
  c[1] = p0 * S.R[1] + p1 * S.R[4] + p2 * S.R[7];
  c[2] = p0 * S.R[2] + p1 * S.R[5] + p2 * S.R[8];
  float* cp = canon + (size_t)pt * 3;
  cp[0] = c[0]; cp[1] = c[1]; cp[2] = c[2];
  for (int e = 0; e < 3; ++e) {
    red[threadIdx.x] = c[e] * c[e] * c[e];
    __syncthreads();
    for (int s = 128; s > 0; s >>= 1) {
      if (threadIdx.x < s) red[threadIdx.x] += red[threadIdx.x + s];
      __syncthreads();
    }
    if (threadIdx.x == 0) atomicAdd(&st[b].skew[e], red[0]);
    __syncthreads();
  }
}

// ------------------------------------------------------------ signs (tiny)
__global__ void k_signs(BatchStats* st) {
  int b = blockIdx.x * blockDim.x + threadIdx.x;
  if (b >= B_DIM) return;
  float s[3]; int flips = 0;
  for (int e = 0; e < 3; ++e) {
    float sk = st[b].skew[e];           // sign(mean) == sign(sum); sign(0)->+1
    s[e] = (sk < 0.0f) ? -1.0f : 1.0f;
    if (sk < 0.0f) flips++;
  }
  if (flips & 1) s[2] = -s[2];
  for (int e = 0; e < 3; ++e) st[b].s[e] = s[e];
}

// -------- pass 5: stable 3-key LSD radix sort, one workgroup per batch.
// Records are {key32 << 32 | idx}. 4-bit digits, 8 passes per dim, dims 2,1,0
// (equivalent to the reference's chained stable argsorts).
__device__ __forceinline__ unsigned int f2key(float v) {
  unsigned int u = __float_as_uint(v);
  return (u & 0x80000000u) ? ~u : (u | 0x80000000u);
}

__global__ void __launch_bounds__(256) k_sort(const float* __restrict__ canon,
                                              const BatchStats* st,
                                              unsigned long long* recA,
                                              unsigned long long* recB) {
  __shared__ unsigned int H[16 * 256];   // per-thread digit histograms / excl scans
  __shared__ unsigned int O[16 * 256];   // running scatter offsets
  __shared__ unsigned int baseS[16];
  const int b = blockIdx.x, tid = threadIdx.x;
  const float* cb = canon + (size_t)b * N_PTS * 3;
  const float sv[3] = {st[b].s[0], st[b].s[1], st[b].s[2]};
  unsigned long long* bufs[2] = {recA + (size_t)b * N_PTS,
                                 recB + (size_t)b * N_PTS};
  int cur = 0;
  for (int dim = 2; dim >= 0; --dim) {
    __syncthreads();
    // (re)build records in current order with this dim's key
    for (int j = tid; j < N_PTS; j += 256) {
      unsigned int idx =
          (dim == 2) ? (unsigned int)j
                     : (unsigned int)(bufs[cur][j] & 0xffffffffu);
      unsigned int u = f2key(cb[(size_t)idx * 3 + dim] * sv[dim]);
      bufs[cur][j] = ((unsigned long long)u << 32) | idx;
    }
    for (int p = 0; p < 8; ++p) {
      __syncthreads();
      for (int d = 0; d < 16; ++d) H[d * 256 + tid] = 0;
      const unsigned long long* src = bufs[cur];
      unsigned long long* dst = bufs[cur ^ 1];
      const int start = tid * 256;                 // contiguous => stable
      const int sh = 32 + 4 * p;
      for (int i = 0; i < 256; ++i) {
        __builtin_prefetch(&src[start + i + 32], 0, 0);  // global_prefetch_b8
        unsigned int d = (unsigned int)(src[start + i] >> sh) & 15u;
        H[d * 256 + tid] += 1;                     // own column: no conflicts
      }
      __syncthreads();
      if (tid < 16) {                              // exclusive scan of row tid
        unsigned int run = 0;
        for (int t = 0; t < 256; ++t) {
          unsigned int v = H[tid * 256 + t];
          H[tid * 256 + t] = run;
          run += v;
        }
        baseS[tid] = run;                          // digit total
      }
      __syncthreads();
      if (tid == 0) {                              // exclusive scan of totals
        unsigned int r = 0;
        for (int d = 0; d < 16; ++d) { unsigned int t = baseS[d]; baseS[d] = r; r += t; }
      }
      __syncthreads();
      for (int d = 0; d < 16; ++d)
        O[d * 256 + tid] = H[d * 256 + tid] + baseS[d];
      __syncthreads();
      for (int i = 0; i < 256; ++i) {
        __builtin_prefetch(&src[start + i + 32], 0, 0);
        unsigned long long r = src[start + i];
        unsigned int d = (unsigned int)(r >> sh) & 15u;
        dst[O[d * 256 + tid]++] = r;
      }
      cur ^= 1;
    }
  }
  // 24 passes total -> cur ends at 0 (recA holds the sorted records)
}

// ------------------------------------------ pass 6: gather canonical2 -> out
__global__ void __launch_bounds__(256) k_gather(const float* __restrict__ canon,
                                                const BatchStats* st,
                                                const unsigned long long* recA,
                                                float* __restrict__ out) {
  const int i = blockIdx.x * blockDim.x + threadIdx.x;   // over B*N
  const int b = i >> 16;
  const unsigned int idx =
      (unsigned int)(recA[(size_t)b * N_PTS + (i & 65535)] & 0xffffffffu);
  const float* cp = canon + ((size_t)b * N_PTS + idx) * 3;
  float* op = out + (size_t)i * 3;
  op[0] = cp[0] * st[b].s[0];
  op[1] = cp[1] * st[b].s[1];
  op[2] = cp[2] * st[b].s[2];
}

// ---------------------------------------------------------------- launcher
extern "C" void kernel_launch(void* const* d_in, const int* in_sizes, int n_in,
                              void* d_out, int out_size, void* d_ws,
                              size_t ws_size, hipStream_t stream) {
  (void)in_sizes; (void)n_in; (void)out_size; (void)ws_size;
  const float* x = (const float*)d_in[0];
  float* out = (float*)d_out;

  char* ws = (char*)d_ws;
  size_t off = 0;
  BatchStats* st = (BatchStats*)(ws + off);
  off += (sizeof(BatchStats) * B_DIM + 255) & ~(size_t)255;
  float* canon = (float*)(ws + off);
  off += ((size_t)B_DIM * N_PTS * 3 * sizeof(float) + 255) & ~(size_t)255;
  unsigned long long* recA = (unsigned long long*)(ws + off);
  off += (size_t)B_DIM * N_PTS * sizeof(unsigned long long);
  unsigned long long* recB = (unsigned long long*)(ws + off);

  const int ptBlocks = (B_DIM * N_PTS) / 256;   // 32768

  k_zero<<<32, 256, 0, stream>>>(st);
  k_moments<<<dim3(64, B_DIM), 256, 0, stream>>>(x, st);
  k_mean<<<1, 128, 0, stream>>>(st);
  k_maxnorm<<<ptBlocks, 256, 0, stream>>>(x, st);
  k_eigen<<<1, 128, 0, stream>>>(st);
  k_transform<<<ptBlocks, 256, 0, stream>>>(x, st, canon);
  k_signs<<<1, 128, 0, stream>>>(st);
  k_sort<<<B_DIM, 256, 0, stream>>>(canon, st, recA, recB);
  k_gather<<<ptBlocks, 256, 0, stream>>>(canon, st, recA, out);
}